// DiscriminativeLoss_56839597195849
// MI455X (gfx1250) — compile-verified
//
#include <hip/hip_runtime.h>

// ---------------------------------------------------------------------------
// DiscriminativeLoss on MI455X (gfx1250, wave32).
// Roofline: 2 streaming passes over 128 MiB data + labels ~= 272 MiB
// -> ~12 us at 23.3 TB/s. Bandwidth-bound; WMMA (f32 16x16x4, exact FMA)
// computes the 32x32 centers Gram matrix for the pairwise-distance term.
// ---------------------------------------------------------------------------

#define CMAX 64          // workspace sized for up to 64 clusters (ref uses 32)
#define DP   33          // padded LDS stride (D=32 + 1) to avoid bank conflicts

static constexpr float DELTA_VAR  = 1.0f;
static constexpr float DELTA_DIST = 2.0f;

// Workspace layout (floats)
enum : int {
  WS_SUMS    = 0,                    // [CMAX*32]  per-cluster feature sums
  WS_COUNTS  = CMAX * 32,            // [CMAX]     per-cluster counts
  WS_VAR     = WS_COUNTS + CMAX,     // [1]        variance-term accumulator
  WS_CENTERS = WS_VAR + 1,           // [CMAX*32]  centers
  WS_TOTAL   = WS_CENTERS + CMAX * 32
};

typedef float v2f __attribute__((ext_vector_type(2)));
typedef float v8f __attribute__((ext_vector_type(8)));

// ---------------------------------------------------------------------------
__global__ void k_zero(float* __restrict__ ws) {
  int i = blockIdx.x * blockDim.x + threadIdx.x;
  if (i < WS_TOTAL) ws[i] = 0.0f;
}

// ---------------------------------------------------------------------------
// Pass 1: per-cluster sums + counts. Each wave (32 lanes) takes 32 consecutive
// points; for each d the wave reads 128 contiguous bytes (fully coalesced).
// LDS-privatized accumulators (stride DP=33), one global-atomic flush / block.
__global__ void k_segsum(const float* __restrict__ data,
                         const int*   __restrict__ lab,
                         const int*   __restrict__ Cptr,
                         float* __restrict__ ws, int N, int D) {
  __shared__ float lsum[CMAX * DP];
  __shared__ float lcnt[CMAX];
  const int C   = *Cptr;
  const int tid = threadIdx.x;
  for (int i = tid; i < CMAX * DP; i += blockDim.x) lsum[i] = 0.0f;
  for (int i = tid; i < CMAX;      i += blockDim.x) lcnt[i] = 0.0f;
  __syncthreads();

  const int lane = tid & 31;
  const int wid  = (blockIdx.x * blockDim.x + tid) >> 5;
  const int nw   = (gridDim.x * blockDim.x) >> 5;
  for (long long n0 = (long long)wid * 32; n0 < N; n0 += (long long)nw * 32) {
    const int n = (int)n0 + lane;
    if (n < N) {
      const int c = lab[n];                       // coalesced per wave
      atomicAdd(&lcnt[c], 1.0f);
      const float* __restrict__ p = data + n;
      for (int d = 0; d < D; ++d)
        atomicAdd(&lsum[c * DP + d], p[(size_t)d * N]);
    }
  }
  __syncthreads();

  for (int i = tid; i < C * D; i += blockDim.x) {
    const int c = i / D, d = i - c * D;
    const float v = lsum[c * DP + d];
    if (v != 0.0f) atomicAdd(&ws[WS_SUMS + i], v);
  }
  for (int c = tid; c < C; c += blockDim.x) {
    const float v = lcnt[c];
    if (v != 0.0f) atomicAdd(&ws[WS_COUNTS + c], v);
  }
}

// ---------------------------------------------------------------------------
__global__ void k_centers(const int* __restrict__ Cptr,
                          float* __restrict__ ws, int D) {
  const int C = *Cptr;
  const int i = blockIdx.x * blockDim.x + threadIdx.x;
  if (i < C * D)
    ws[WS_CENTERS + i] = ws[WS_SUMS + i] / ws[WS_COUNTS + i / D];
}

// ---------------------------------------------------------------------------
// Pass 2: variance term. Centers cached in LDS (padded stride); same coalesced
// streaming pattern; per-lane squared-distance accumulation in registers.
__global__ void k_var(const float* __restrict__ data,
                      const int*   __restrict__ lab,
                      const int*   __restrict__ Cptr,
                      float* __restrict__ ws, int N, int D) {
  __shared__ float lc[CMAX * DP];
  __shared__ float red[8];
  const int C = *Cptr;
  for (int i = threadIdx.x; i < C * D; i += blockDim.x) {
    const int c = i / D, d = i - c * D;
    lc[c * DP + d] = ws[WS_CENTERS + i];
  }
  __syncthreads();

  const int lane  = threadIdx.x & 31;
  const int wslot = threadIdx.x >> 5;
  const int wid   = (blockIdx.x * blockDim.x + threadIdx.x) >> 5;
  const int nw    = (gridDim.x * blockDim.x) >> 5;
  float acc = 0.0f;
  for (long long n0 = (long long)wid * 32; n0 < N; n0 += (long long)nw * 32) {
    const int n = (int)n0 + lane;
    if (n < N) {
      const int c = lab[n];
      const float* __restrict__ ctr = &lc[c * DP];
      const float* __restrict__ p   = data + n;
      float s = 0.0f;
      for (int d = 0; d < D; ++d) {
        const float v = p[(size_t)d * N] - ctr[d];
        s += v * v;
      }
      float h = fmaxf(sqrtf(s) - DELTA_VAR, 0.0f);
      acc += h * h;
    }
  }
  for (int off = 16; off > 0; off >>= 1) acc += __shfl_down(acc, off, 32);
  if (lane == 0) red[wslot] = acc;
  __syncthreads();
  if (threadIdx.x == 0) {
    float s = 0.0f;
    const int nwb = blockDim.x >> 5;
    for (int w = 0; w < nwb; ++w) s += red[w];
    atomicAdd(&ws[WS_VAR], s);
  }
}

// ---------------------------------------------------------------------------
// Finalize: one wave (32 threads, EXEC all-ones -> WMMA legal).
// Gram G = centers * centers^T via V_WMMA_F32_16X16X4_F32 (exact f32 FMA):
// 2x2 tiles of 16x16, K = 32 in 8 steps of 4. ISA 7.12.2 f32 A-layout:
// lanes 0-15 hold (K=0,K=1), lanes 16-31 hold (K=2,K=3) per K-step; B =
// centers^T has the symmetric layout, so A/B fragments share register values.
__global__ void k_final(const int* __restrict__ Cptr,
                        const float* __restrict__ ws,
                        float* __restrict__ out, int D) {
  const int C = *Cptr;                         // uniform scalar load
  const float* __restrict__ centers = ws + WS_CENTERS;
  __shared__ float ln2[CMAX];
  const int lane = threadIdx.x;                // blockDim.x == 32

  for (int c = lane; c < C; c += 32) {         // per-center squared norms
    float s = 0.0f;
    for (int d = 0; d < D; ++d) { const float v = centers[c * D + d]; s += v * v; }
    ln2[c] = s;
  }
  __syncthreads();

  float dist_sum = 0.0f;
  if (C == 32 && D == 32) {                    // uniform branch: WMMA path
    v8f acc00 = {}, acc01 = {}, acc10 = {}, acc11 = {};
    const int m  = lane & 15;
    const int kh = (lane >> 4) * 2;
#pragma unroll
    for (int kk = 0; kk < 8; ++kk) {
      const int k = kk * 4 + kh;
      v2f f0, f1;
      f0.x = centers[m * 32 + k];
      f0.y = centers[m * 32 + k + 1];
      f1.x = centers[(16 + m) * 32 + k];
      f1.y = centers[(16 + m) * 32 + k + 1];
      acc00 = __builtin_amdgcn_wmma_f32_16x16x4_f32(false, f0, false, f0, (short)0, acc00, false, false);
      acc01 = __builtin_amdgcn_wmma_f32_16x16x4_f32(false, f0, false, f1, (short)0, acc01, false, false);
      acc10 = __builtin_amdgcn_wmma_f32_16x16x4_f32(false, f1, false, f0, (short)0, acc10, false, false);
      acc11 = __builtin_amdgcn_wmma_f32_16x16x4_f32(false, f1, false, f1, (short)0, acc11, false, false);
    }
    const v8f accs[4] = {acc00, acc01, acc10, acc11};
    const int jbase = lane & 15;
    const int rbase = (lane >> 4) * 8;
#pragma unroll
    for (int t = 0; t < 4; ++t) {
      const int ti = t >> 1, tj = t & 1;
#pragma unroll
      for (int r = 0; r < 8; ++r) {
        const int i = ti * 16 + rbase + r;
        const int j = tj * 16 + jbase;
        const float g   = accs[t][r];
        const float sq  = ln2[i] + ln2[j] - 2.0f * g;
        const float eye = (i == j) ? 1.0f : 0.0f;
        const float cd  = sqrtf(sq + eye);
        const float h   = fmaxf(2.0f * DELTA_DIST - cd, 0.0f);
        dist_sum += h * h * (1.0f - eye);
      }
    }
  } else {                                     // generic fallback (never for ref)
    for (int i = lane; i < C; i += 32) {
      for (int j = 0; j < C; ++j) {
        float g = 0.0f;
        for (int d = 0; d < D; ++d) g += centers[i * D + d] * centers[j * D + d];
        const float sq  = ln2[i] + ln2[j] - 2.0f * g;
        const float eye = (i == j) ? 1.0f : 0.0f;
        const float cd  = sqrtf(sq + eye);
        const float h   = fmaxf(2.0f * DELTA_DIST - cd, 0.0f);
        dist_sum += h * h * (1.0f - eye);
      }
    }
  }

  float reg = 0.0f;
  const float dr = sqrtf((float)D);
  for (int c = lane; c < C; c += 32)
    reg += fmaxf(sqrtf(ln2[c]) - dr, 0.0f);

  for (int off = 16; off > 0; off >>= 1) {
    dist_sum += __shfl_down(dist_sum, off, 32);
    reg      += __shfl_down(reg, off, 32);
  }
  if (lane == 0) {
    const float Cf = (float)C;
    out[0] = ws[WS_VAR] / Cf + dist_sum / (Cf * (Cf - 1.0f)) + reg / Cf;
  }
}

// ---------------------------------------------------------------------------
extern "C" void kernel_launch(void* const* d_in, const int* in_sizes, int n_in,
                              void* d_out, int out_size, void* d_ws, size_t ws_size,
                              hipStream_t stream) {
  const float* data = (const float*)d_in[0];   // (D, H, W) channel-major
  const int*   lab  = (const int*)d_in[1];     // (H, W)
  const int*   Cptr = (const int*)d_in[2];     // scalar C on device
  const int N = in_sizes[1];                   // H*W
  const int D = in_sizes[0] / N;               // 32
  float* ws  = (float*)d_ws;
  float* out = (float*)d_out;

  const int STREAM_BLOCKS = 512;               // 4K waves: saturates HBM streams

  k_zero   <<<(WS_TOTAL + 255) / 256, 256, 0, stream>>>(ws);
  k_segsum <<<STREAM_BLOCKS, 256, 0, stream>>>(data, lab, Cptr, ws, N, D);
  k_centers<<<(CMAX * 32 + 255) / 256, 256, 0, stream>>>(Cptr, ws, D);
  k_var    <<<STREAM_BLOCKS, 256, 0, stream>>>(data, lab, Cptr, ws, N, D);
  k_final  <<<1, 32, 0, stream>>>(Cptr, ws, out, D);
}